// S4DRNetv2_v2_65558380806641
// MI455X (gfx1250) — compile-verified
//
#include <hip/hip_runtime.h>
#include <hip/hip_bf16.h>
#include <cstdint>

#define B_   8
#define N_   4096
#define P_   27
#define KNN_ 20
#define NEG_BIG -3.402823466e38f

typedef __attribute__((ext_vector_type(2))) float v2f;
typedef __attribute__((ext_vector_type(8))) float v8f;

__device__ __forceinline__ float lrelu(float x) { return x >= 0.f ? x : 0.2f * x; }

#if defined(__AMDGCN__) && __has_builtin(__builtin_amdgcn_wmma_f32_16x16x4_f32)
#define HAVE_WMMA_F32X4 1
#endif

// D = A(16x4) * B(4x16) + C, f32, wave32.  A: lanes0-15 row m, {K0,K1}; lanes16-31 row m, {K2,K3}.
// B: lanes0-15 col n, {K0,K1}; lanes16-31 col n, {K2,K3}.  C/D: vgpr v -> M=v(+8 for hi half), N=lane&15.
__device__ __forceinline__ v8f wmma_f32_k4(v2f A, v2f Bv, v8f C) {
#ifdef HAVE_WMMA_F32X4
  return __builtin_amdgcn_wmma_f32_16x16x4_f32(false, A, false, Bv, (short)0, C, false, false);
#else
  int lane = (int)__lane_id();
  int n = lane & 15, half = lane >> 4;
  float b0 = __shfl(Bv.x, n, 32),      b1 = __shfl(Bv.y, n, 32);
  float b2 = __shfl(Bv.x, n + 16, 32), b3 = __shfl(Bv.y, n + 16, 32);
#pragma unroll
  for (int v = 0; v < 8; v++) {
    int m = v + half * 8;
    float a0 = __shfl(A.x, m, 32),      a1 = __shfl(A.y, m, 32);
    float a2 = __shfl(A.x, m + 16, 32), a3 = __shfl(A.y, m + 16, 32);
    C[v] += a0 * b0 + a1 * b1 + a2 * b2 + a3 * b3;
  }
  return C;
#endif
}

__device__ __forceinline__ void atomicMaxFloat(float* addr, float val) {
  if (val >= 0.f) atomicMax((int*)addr, __float_as_int(val));
  else            atomicMin((unsigned int*)addr, __float_as_uint(val));
}

// ---------------------------------------------------------------- utility kernels
__global__ void fill_kernel(float* p, float v, int n) {
  int i = blockIdx.x * blockDim.x + threadIdx.x;
  if (i < n) p[i] = v;
}

__global__ void xx_kernel(const float* __restrict__ x, float* __restrict__ xx) {
  int i = blockIdx.x * blockDim.x + threadIdx.x;
  if (i >= B_ * N_) return;
  int n = i % N_, b = i / N_;
  const float* xb = x + ((size_t)b * 9 + 6) * N_ + n;
  float a = xb[0], c = xb[N_], d = xb[2 * N_];
  xx[i] = a * a + c * c + d * d;
}

__global__ void hist_kernel(const float* __restrict__ x, const int* __restrict__ p2v,
                            float* cnt, float* psum) {
  int i = blockIdx.x * blockDim.x + threadIdx.x;
  if (i >= B_ * N_) return;
  int n = i % N_, b = i / N_;
  int p = p2v[i];
  atomicAdd(&cnt[b * P_ + p], 1.0f);
#pragma unroll
  for (int c = 0; c < 3; c++)
    atomicAdd(&psum[(b * 3 + c) * P_ + p], x[((size_t)b * 9 + 6 + c) * N_ + n]);
}

__global__ void center_kernel(const float* psum, const float* cnt, float* center) {
  int i = blockIdx.x * blockDim.x + threadIdx.x;
  if (i >= B_ * 3 * P_) return;
  int p = i % P_, b = i / (3 * P_);
  center[i] = psum[i] / fmaxf(cnt[b * P_ + p], 1.0f);
}

// y[b,o,pos] = sum_c W[o,c]*x[b,c,pos]; accumulates per-channel sum/sumsq for train-BN
__global__ void conv_stats_kernel(const float* __restrict__ W, const float* __restrict__ xin,
                                  float* __restrict__ y, float* stats,
                                  int B, int Cin, int Cout, int Npos) {
  int i = blockIdx.x * blockDim.x + threadIdx.x;
  int total = B * Cout * Npos;
  if (i >= total) return;
  int pos = i % Npos, o = (i / Npos) % Cout, b = i / (Npos * Cout);
  const float* xb = xin + ((size_t)b * Cin) * Npos + pos;
  const float* wr = W + (size_t)o * Cin;
  float acc = 0.f;
  for (int c = 0; c < Cin; c++) acc += wr[c] * xb[(size_t)c * Npos];
  y[i] = acc;
  atomicAdd(&stats[o], acc);
  atomicAdd(&stats[Cout + o], acc * acc);
}

__global__ void bn_finalize_kernel(const float* stats, const float* g, const float* beta,
                                   float* ab, int Cout, float invM) {
  int c = blockIdx.x * blockDim.x + threadIdx.x;
  if (c >= Cout) return;
  float mean = stats[c] * invM;
  float var = stats[Cout + c] * invM - mean * mean;
  float a = g[c] * rsqrtf(var + 1e-5f);
  ab[c] = a;
  ab[Cout + c] = beta[c] - mean * a;
}

__global__ void bn_lrelu_apply_kernel(float* y, const float* ab, int Cout, int Npos, int total) {
  int i = blockIdx.x * blockDim.x + threadIdx.x;
  if (i >= total) return;
  int c = (i / Npos) % Cout;
  y[i] = lrelu(ab[c] * y[i] + ab[Cout + c]);
}

// ---------------------------------------------------------------- kNN: WMMA 16x16x4 f32 tiles
__global__ void __launch_bounds__(512) knn_kernel(const float* __restrict__ x,
                                                  const float* __restrict__ xx,
                                                  int* __restrict__ idx) {
  extern __shared__ float dist[];  // [16][N_]
  int b = blockIdx.x / (N_ / 16);
  int i0 = (blockIdx.x % (N_ / 16)) * 16;
  int lane = threadIdx.x & 31, wave = threadIdx.x >> 5;
  int half = lane >> 4, l = lane & 15;
  const float* X0 = x + ((size_t)b * 9 + 6) * N_;
  const float* X1 = X0 + N_;
  const float* X2 = X1 + N_;

  v2f A;
  A.x = half ? X2[i0 + l] : X0[i0 + l];
  A.y = half ? 0.f : X1[i0 + l];
  float xxi[8];
#pragma unroll
  for (int v = 0; v < 8; v++) xxi[v] = xx[b * N_ + i0 + v + half * 8];

  for (int T = wave; T < N_ / 16; T += 16) {
    int j0 = T * 16;
    v2f Bv;
    Bv.x = half ? X2[j0 + l] : X0[j0 + l];
    Bv.y = half ? 0.f : X1[j0 + l];
    v8f acc = {0.f, 0.f, 0.f, 0.f, 0.f, 0.f, 0.f, 0.f};
    acc = wmma_f32_k4(A, Bv, acc);
    int col = j0 + l;
    float xxc = xx[b * N_ + col];
#pragma unroll
    for (int v = 0; v < 8; v++) {
      int m = v + half * 8;
      dist[m * N_ + col] = 2.0f * acc[v] - xxi[v] - xxc;
    }
  }
  __syncthreads();

  {  // wave r: top-20 of row r
    int r = wave;
    float* row = dist + (size_t)r * N_;
    for (int pass = 0; pass < KNN_; pass++) {
      float best = NEG_BIG;
      int bi = 0;
      for (int e = lane; e < N_; e += 32) {
        float v = row[e];
        if (v > best) { best = v; bi = e; }
      }
#pragma unroll
      for (int off = 16; off; off >>= 1) {
        float ov = __shfl_xor(best, off, 32);
        int oi = __shfl_xor(bi, off, 32);
        if (ov > best || (ov == best && oi < bi)) { best = ov; bi = oi; }
      }
      if (lane == 0) {
        idx[((size_t)b * N_ + i0 + r) * KNN_ + pass] = bi;
        row[bi] = NEG_BIG;
      }
      __syncthreads();  // keep waves in lock-step; also orders the LDS write
    }
  }
}

// ---------------------------------------------------------------- graph-feature convs (WMMA)
__device__ __forceinline__ void build_h0(float* h0, const float* __restrict__ x,
                                         const int* __restrict__ idx, int b, int p0, int lane) {
  for (int t = lane; t < 20 * 16; t += 32) {
    int k = t >> 4, pi = t & 15;
    float val = 0.f;
    if (k < 18) {
      int p = p0 + pi;
      int n = p / KNN_, kk = p - n * KNN_;
      if (k < 9) {
        int j = idx[((size_t)b * N_ + n) * KNN_ + kk];
        const float* xc = x + ((size_t)b * 9 + k) * N_;
        val = xc[j] - xc[n];
      } else {
        val = x[((size_t)b * 9 + (k - 9)) * N_ + n];
      }
    }
    h0[t] = val;
  }
}

__device__ __forceinline__ void gemm18(const float* __restrict__ W, const float* h0, int lane,
                                       v8f acc[4]) {
  int half = lane >> 4, l = lane & 15;
#pragma unroll
  for (int k0 = 0; k0 < 20; k0 += 4) {
    int ka = k0 + half * 2;
    v2f Bv;
    Bv.x = h0[ka * 16 + l];
    Bv.y = h0[(ka + 1) * 16 + l];
#pragma unroll
    for (int mt = 0; mt < 4; mt++) {
      int row = mt * 16 + l;
      v2f Av;
      Av.x = (ka < 18) ? W[row * 18 + ka] : 0.f;
      Av.y = (ka + 1 < 18) ? W[row * 18 + ka + 1] : 0.f;
      acc[mt] = wmma_f32_k4(Av, Bv, acc[mt]);
    }
  }
}

__device__ __forceinline__ void gemm64(const float* __restrict__ W, const float* h1, int lane,
                                       v8f acc[4]) {
  int half = lane >> 4, l = lane & 15;
  for (int k0 = 0; k0 < 64; k0 += 4) {
    int ka = k0 + half * 2;
    v2f Bv;
    Bv.x = h1[ka * 16 + l];
    Bv.y = h1[(ka + 1) * 16 + l];
#pragma unroll
    for (int mt = 0; mt < 4; mt++) {
      int row = mt * 16 + l;
      v2f Av;
      Av.x = W[row * 64 + ka];
      Av.y = W[row * 64 + ka + 1];
      acc[mt] = wmma_f32_k4(Av, Bv, acc[mt]);
    }
  }
}

__device__ __forceinline__ void stats_reduce(v8f acc[4], float* stats, int lane) {
  int half = lane >> 4;
#pragma unroll
  for (int mt = 0; mt < 4; mt++) {
#pragma unroll
    for (int v = 0; v < 8; v++) {
      float s = acc[mt][v];
      float q = s * s;
#pragma unroll
      for (int off = 1; off < 16; off <<= 1) {
        s += __shfl_xor(s, off, 32);
        q += __shfl_xor(q, off, 32);
      }
      if ((lane & 15) == 0) {
        int ch = mt * 16 + v + half * 8;
        atomicAdd(&stats[ch], s);
        atomicAdd(&stats[64 + ch], q);
      }
    }
  }
}

__global__ void __launch_bounds__(128) gf1_kernel(const float* __restrict__ x,
                                                  const int* __restrict__ idx,
                                                  const float* __restrict__ w1, float* stats) {
  __shared__ float h0s[4][320];
  int lane = threadIdx.x & 31, wv = threadIdx.x >> 5;
  int tile = blockIdx.x * 4 + wv;
  int b = tile / (N_ * KNN_ / 16);
  int p0 = (tile % (N_ * KNN_ / 16)) * 16;
  build_h0(h0s[wv], x, idx, b, p0, lane);
  __syncthreads();
  v8f z = {0.f, 0.f, 0.f, 0.f, 0.f, 0.f, 0.f, 0.f};
  v8f acc[4] = {z, z, z, z};
  gemm18(w1, h0s[wv], lane, acc);
  stats_reduce(acc, stats, lane);
}

__global__ void __launch_bounds__(128) gf2_kernel(const float* __restrict__ x,
                                                  const int* __restrict__ idx,
                                                  const float* __restrict__ w1,
                                                  const float* __restrict__ w2,
                                                  const float* __restrict__ ab1, float* stats) {
  __shared__ float h0s[4][320];
  __shared__ float h1s[4][1024];
  int lane = threadIdx.x & 31, wv = threadIdx.x >> 5;
  int tile = blockIdx.x * 4 + wv;
  int b = tile / (N_ * KNN_ / 16);
  int p0 = (tile % (N_ * KNN_ / 16)) * 16;
  build_h0(h0s[wv], x, idx, b, p0, lane);
  __syncthreads();
  v8f z = {0.f, 0.f, 0.f, 0.f, 0.f, 0.f, 0.f, 0.f};
  v8f acc[4] = {z, z, z, z};
  gemm18(w1, h0s[wv], lane, acc);
  int half = lane >> 4, l = lane & 15;
#pragma unroll
  for (int mt = 0; mt < 4; mt++)
#pragma unroll
    for (int v = 0; v < 8; v++) {
      int ch = mt * 16 + v + half * 8;
      h1s[wv][ch * 16 + l] = lrelu(ab1[ch] * acc[mt][v] + ab1[64 + ch]);
    }
  __syncthreads();
  v8f acc2[4] = {z, z, z, z};
  gemm64(w2, h1s[wv], lane, acc2);
  stats_reduce(acc2, stats, lane);
}

__global__ void __launch_bounds__(128) gf3_kernel(const float* __restrict__ x,
                                                  const int* __restrict__ idx,
                                                  const float* __restrict__ w1,
                                                  const float* __restrict__ w2,
                                                  const float* __restrict__ ab1,
                                                  const float* __restrict__ ab2,
                                                  float* __restrict__ x1) {
  __shared__ float h0s[4][320];
  __shared__ float h1s[4][1024];
  int lane = threadIdx.x & 31, wv = threadIdx.x >> 5;
  int tile = blockIdx.x * 4 + wv;
  int b = tile / (N_ * KNN_ / 16);
  int p0 = (tile % (N_ * KNN_ / 16)) * 16;
  build_h0(h0s[wv], x, idx, b, p0, lane);
  __syncthreads();
  v8f z = {0.f, 0.f, 0.f, 0.f, 0.f, 0.f, 0.f, 0.f};
  v8f acc[4] = {z, z, z, z};
  gemm18(w1, h0s[wv], lane, acc);
  int half = lane >> 4, l = lane & 15;
#pragma unroll
  for (int mt = 0; mt < 4; mt++)
#pragma unroll
    for (int v = 0; v < 8; v++) {
      int ch = mt * 16 + v + half * 8;
      h1s[wv][ch * 16 + l] = lrelu(ab1[ch] * acc[mt][v] + ab1[64 + ch]);
    }
  __syncthreads();
  v8f acc2[4] = {z, z, z, z};
  gemm64(w2, h1s[wv], lane, acc2);
  int p = p0 + l;
  int n = p / KNN_;
#pragma unroll
  for (int mt = 0; mt < 4; mt++)
#pragma unroll
    for (int v = 0; v < 8; v++) {
      int ch = mt * 16 + v + half * 8;
      float h2 = lrelu(ab2[ch] * acc2[mt][v] + ab2[64 + ch]);
      atomicMaxFloat(&x1[((size_t)b * 64 + ch) * N_ + n], h2);
    }
}

// ---------------------------------------------------------------- part pooling / SSDP
__global__ void pool_x1_kernel(const float* __restrict__ x1, const int* __restrict__ p2v,
                               float* psum) {
  int i = blockIdx.x * blockDim.x + threadIdx.x;
  if (i >= B_ * 64 * N_) return;
  int n = i % N_, c = (i / N_) % 64, b = i / (64 * N_);
  int p = p2v[b * N_ + n];
  atomicAdd(&psum[(b * 64 + c) * P_ + p], x1[i]);
}

__global__ void pf1_kernel(const float* pce, const float* psum, const float* cnt, float* pf1) {
  int i = blockIdx.x * blockDim.x + threadIdx.x;
  if (i >= B_ * 64 * P_) return;
  int p = i % P_, b = i / (64 * P_);
  pf1[i] = pce[i] + psum[i] / fmaxf(cnt[b * P_ + p], 1.0f);
}

__global__ void e0_kernel(const float* __restrict__ pf, float* __restrict__ e0) {
  int i = blockIdx.x * blockDim.x + threadIdx.x;
  if (i >= B_ * 128 * P_ * P_) return;
  int j = i % P_;
  int ii = (i / P_) % P_;
  int c = (i / (P_ * P_)) % 128;
  int b = i / (128 * P_ * P_);
  const float* pb = pf + (size_t)b * 64 * P_;
  e0[i] = (c < 64) ? (pb[c * P_ + j] - pb[c * P_ + ii]) : pb[(c - 64) * P_ + ii];
}

__global__ void __launch_bounds__(128) attn1_kernel(const float* __restrict__ y,
                                                    const float* __restrict__ ab,
                                                    const float* __restrict__ at,
                                                    float* __restrict__ e1p) {
  __shared__ float e1[64 * P_];
  __shared__ float lg[4 * P_];
  __shared__ float aw[4 * P_];
  int b = blockIdx.x / P_, n = blockIdx.x % P_;
  size_t base = ((size_t)b * 64) * (P_ * P_) + n * P_;
  for (int t = threadIdx.x; t < 64 * P_; t += blockDim.x) {
    int c = t / P_, k = t % P_;
    e1[t] = lrelu(ab[c] * y[base + (size_t)c * (P_ * P_) + k] + ab[64 + c]);
  }
  __syncthreads();
  for (int t = threadIdx.x; t < 4 * P_; t += blockDim.x) {
    int h = t / P_, k = t % P_;
    float s = 0.f;
    for (int d = 0; d < 16; d++) s += at[d] * lrelu(e1[(d * 4 + h) * P_ + k]);
    lg[t] = s;
  }
  __syncthreads();
  if (threadIdx.x < 4) {
    int h = threadIdx.x;
    float m = NEG_BIG;
    for (int k = 0; k < P_; k++) m = fmaxf(m, lg[h * P_ + k]);
    float s = 0.f;
    for (int k = 0; k < P_; k++) { float e = expf(lg[h * P_ + k] - m); aw[h * P_ + k] = e; s += e; }
    float inv = 1.f / s;
    for (int k = 0; k < P_; k++) aw[h * P_ + k] *= inv;
  }
  __syncthreads();
  for (int t = threadIdx.x; t < 64 * P_; t += blockDim.x) {
    int c = t / P_, k = t % P_, h = c & 3;
    e1p[base + (size_t)c * (P_ * P_) + k] = aw[h * P_ + k] * e1[t];
  }
}

__global__ void __launch_bounds__(128) attn2_kernel(const float* __restrict__ y2,
                                                    const float* __restrict__ ab,
                                                    const float* __restrict__ at,
                                                    const float* __restrict__ pd,
                                                    float* __restrict__ gout) {
  __shared__ float e2[64 * P_];
  __shared__ float lg[4 * P_];
  __shared__ float aw[4 * P_];
  __shared__ float gh[P_];
  int b = blockIdx.x / P_, n = blockIdx.x % P_;
  size_t base = ((size_t)b * 64) * (P_ * P_) + n * P_;
  for (int t = threadIdx.x; t < 64 * P_; t += blockDim.x) {
    int c = t / P_, k = t % P_;
    e2[t] = lrelu(ab[c] * y2[base + (size_t)c * (P_ * P_) + k] + ab[64 + c]);
  }
  if (threadIdx.x < P_) {
    float d = pd[((size_t)b * P_ + n) * P_ + threadIdx.x];
    gh[threadIdx.x] = expf(-0.5f * d * d) * 0.3989422804014327f;
  }
  __syncthreads();
  for (int t = threadIdx.x; t < 4 * P_; t += blockDim.x) {
    int h = t / P_, k = t % P_;
    float s = 0.f;
    for (int d = 0; d < 16; d++) s += at[d] * lrelu(gh[k] * e2[(d * 4 + h) * P_ + k]);
    lg[t] = s;
  }
  __syncthreads();
  if (threadIdx.x < 4) {
    int h = threadIdx.x;
    float m = NEG_BIG;
    for (int k = 0; k < P_; k++) m = fmaxf(m, lg[h * P_ + k]);
    float s = 0.f;
    for (int k = 0; k < P_; k++) { float e = expf(lg[h * P_ + k] - m); aw[h * P_ + k] = e; s += e; }
    float inv = 1.f / s;
    for (int k = 0; k < P_; k++) aw[h * P_ + k] *= inv;
  }
  __syncthreads();
  for (int c = threadIdx.x; c < 64; c += blockDim.x) {
    int h = c & 3;
    float s = 0.f;
    for (int k = 0; k < P_; k++) s += aw[h * P_ + k] * e2[c * P_ + k];
    gout[((size_t)b * 64 + c) * P_ + n] += s * (1.f / 27.f);
  }
}

__global__ void part2point_kernel(const float* __restrict__ gout, const int* __restrict__ p2v,
                                  float* __restrict__ out) {
  int i = blockIdx.x * blockDim.x + threadIdx.x;
  if (i >= B_ * 64 * N_) return;
  int n = i % N_, c = (i / N_) % 64, b = i / (64 * N_);
  out[i] = gout[(b * 64 + c) * P_ + p2v[b * N_ + n]];
}

// ---------------------------------------------------------------- workspace layout (floats)
constexpr size_t OFF_STATS_PCE  = 0;       // 128
constexpr size_t OFF_STATS_C1   = 128;
constexpr size_t OFF_STATS_C2   = 256;
constexpr size_t OFF_STATS_EC1A = 384;
constexpr size_t OFF_STATS_EC1B = 512;
constexpr size_t OFF_STATS_HC1A = 640;     // 64
constexpr size_t OFF_STATS_HC1B = 704;
constexpr size_t OFF_STATS_HC2A = 768;     // 16
constexpr size_t OFF_STATS_HC2B = 784;
constexpr size_t OFF_STATS_EC2A = 800;     // 128
constexpr size_t OFF_STATS_EC2B = 928;
constexpr size_t OFF_CNT        = 1056;    // 216
constexpr size_t OFF_PSUM_XYZ   = 1272;    // 648
constexpr size_t OFF_PSUM_X1    = 1920;    // 13824
constexpr size_t OFF_GOUT       = 15744;   // 13824
constexpr size_t ZERO_TOTAL     = 29568;

constexpr size_t OFF_AB_PCE  = 29568;
constexpr size_t OFF_AB_C1   = 29696;
constexpr size_t OFF_AB_C2   = 29824;
constexpr size_t OFF_AB_EC1A = 29952;
constexpr size_t OFF_AB_EC1B = 30080;
constexpr size_t OFF_AB_HC1A = 30208;
constexpr size_t OFF_AB_HC1B = 30272;
constexpr size_t OFF_AB_HC2A = 30336;
constexpr size_t OFF_AB_HC2B = 30352;
constexpr size_t OFF_AB_EC2A = 30368;
constexpr size_t OFF_AB_EC2B = 30496;
constexpr size_t OFF_XX      = 30720;      // 32768
constexpr size_t OFF_CENTER  = 63488;      // 648
constexpr size_t OFF_PCE     = 64256;      // 13824
constexpr size_t OFF_PF1     = 78080;      // 13824
constexpr size_t OFF_IDX     = 92160;      // 655360 (ints)
constexpr size_t OFF_X1      = 747520;     // 2097152
constexpr size_t OFF_E0      = 2844672;    // 746496
constexpr size_t OFF_Y1      = 3591168;    // 373248
constexpr size_t OFF_E1P     = 3964416;    // 373248
constexpr size_t OFF_T       = 4337664;    // 186624
constexpr size_t OFF_Y2      = 4524288;    // 373248  (end 4897536 floats ~ 18.7 MB)

extern "C" void kernel_launch(void* const* d_in, const int* in_sizes, int n_in,
                              void* d_out, int out_size, void* d_ws, size_t ws_size,
                              hipStream_t stream) {
  (void)in_sizes; (void)n_in; (void)out_size; (void)ws_size;
  const float* x    = (const float*)d_in[0];
  const int*   p2v  = (const int*)d_in[1];
  const float* pd   = (const float*)d_in[3];
  const float* pdsp = (const float*)d_in[4];
  const float* pce_w = (const float*)d_in[5];
  const float* c1_w  = (const float*)d_in[6];
  const float* c2_w  = (const float*)d_in[7];
  const float* ec1_w = (const float*)d_in[8];
  const float* ec1sp_w = (const float*)d_in[9];
  const float* hc1_w = (const float*)d_in[10];
  const float* hc2_w = (const float*)d_in[11];
  const float* hc1sp_w = (const float*)d_in[12];
  const float* hc2sp_w = (const float*)d_in[13];
  const float* ec2_w = (const float*)d_in[14];
  const float* ec2sp_w = (const float*)d_in[15];
  const float* at1_w = (const float*)d_in[16];
  const float* at1sp_w = (const float*)d_in[17];
  const float* at2_w = (const float*)d_in[18];
  const float* at2sp_w = (const float*)d_in[19];
  const float* pce_g = (const float*)d_in[20]; const float* pce_b = (const float*)d_in[21];
  const float* c1_g  = (const float*)d_in[22]; const float* c1_b  = (const float*)d_in[23];
  const float* c2_g  = (const float*)d_in[24]; const float* c2_b  = (const float*)d_in[25];
  const float* ec1_g = (const float*)d_in[26]; const float* ec1_b = (const float*)d_in[27];
  const float* ec1sp_g = (const float*)d_in[28]; const float* ec1sp_b = (const float*)d_in[29];
  const float* hc1_g = (const float*)d_in[30]; const float* hc1_b = (const float*)d_in[31];
  const float* hc2_g = (const float*)d_in[32]; const float* hc2_b = (const float*)d_in[33];
  const float* hc1sp_g = (const float*)d_in[34]; const float* hc1sp_b = (const float*)d_in[35];
  const float* hc2sp_g = (const float*)d_in[36]; const float* hc2sp_b = (const float*)d_in[37];
  const float* ec2_g = (const float*)d_in[38]; const float* ec2_b = (const float*)d_in[39];
  const float* ec2sp_g = (const float*)d_in[40]; const float* ec2sp_b = (const float*)d_in[41];

  float* ws = (float*)d_ws;
  float* out0 = (float*)d_out;                       // (B,64,N)
  float* out_hl1   = out0 + (size_t)B_ * 64 * N_;    // (B,5,27,27)
  float* out_hl1sp = out_hl1 + (size_t)B_ * 5 * P_ * P_;

  float* xxb = ws + OFF_XX;   float* center = ws + OFF_CENTER;
  float* pce = ws + OFF_PCE;  float* pf1 = ws + OFF_PF1;
  int* idx = (int*)(ws + OFF_IDX);
  float* x1 = ws + OFF_X1;    float* e0 = ws + OFF_E0;
  float* y1buf = ws + OFF_Y1; float* e1p = ws + OFF_E1P;
  float* tbuf = ws + OFF_T;   float* y2buf = ws + OFF_Y2;
  float* gout = ws + OFF_GOUT;

  // 0. init accumulators
  fill_kernel<<<(ZERO_TOTAL + 255) / 256, 256, 0, stream>>>(ws, 0.0f, (int)ZERO_TOTAL);
  fill_kernel<<<8192, 256, 0, stream>>>(x1, NEG_BIG, B_ * 64 * N_);

  // 1. point->part pooling prereqs + pce branch
  xx_kernel<<<128, 256, 0, stream>>>(x, xxb);
  hist_kernel<<<128, 256, 0, stream>>>(x, p2v, ws + OFF_CNT, ws + OFF_PSUM_XYZ);
  center_kernel<<<3, 256, 0, stream>>>(ws + OFF_PSUM_XYZ, ws + OFF_CNT, center);
  conv_stats_kernel<<<54, 256, 0, stream>>>(pce_w, center, pce, ws + OFF_STATS_PCE, B_, 3, 64, P_);
  bn_finalize_kernel<<<1, 64, 0, stream>>>(ws + OFF_STATS_PCE, pce_g, pce_b, ws + OFF_AB_PCE, 64,
                                           1.0f / (B_ * P_));
  bn_lrelu_apply_kernel<<<54, 256, 0, stream>>>(pce, ws + OFF_AB_PCE, 64, P_, B_ * 64 * P_);

  // 2. kNN (WMMA f32 16x16x4 distance tiles + per-row top-20)
  knn_kernel<<<B_ * (N_ / 16), 512, 16 * N_ * sizeof(float), stream>>>(x, xxb, idx);

  // 3. edge convs c1/c2 with train-BN (3-pass recompute, WMMA GEMMs)
  const float invM = 1.0f / (float)(B_ * N_ * KNN_);
  gf1_kernel<<<10240, 128, 0, stream>>>(x, idx, c1_w, ws + OFF_STATS_C1);
  bn_finalize_kernel<<<1, 64, 0, stream>>>(ws + OFF_STATS_C1, c1_g, c1_b, ws + OFF_AB_C1, 64, invM);
  gf2_kernel<<<10240, 128, 0, stream>>>(x, idx, c1_w, c2_w, ws + OFF_AB_C1, ws + OFF_STATS_C2);
  bn_finalize_kernel<<<1, 64, 0, stream>>>(ws + OFF_STATS_C2, c2_g, c2_b, ws + OFF_AB_C2, 64, invM);
  gf3_kernel<<<10240, 128, 0, stream>>>(x, idx, c1_w, c2_w, ws + OFF_AB_C1, ws + OFF_AB_C2, x1);

  // 4. pool to parts, pf1, e0
  pool_x1_kernel<<<8192, 256, 0, stream>>>(x1, p2v, ws + OFF_PSUM_X1);
  pf1_kernel<<<54, 256, 0, stream>>>(pce, ws + OFF_PSUM_X1, ws + OFF_CNT, pf1);
  e0_kernel<<<2916, 256, 0, stream>>>(pf1, e0);

  // 5. SSDP branches
  const float invS = 1.0f / (float)(B_ * P_ * P_);
  auto run_branch = [&](const float* ec1w, const float* ec1g, const float* ec1b, const float* at1,
                        const float* hc1w, const float* hc1g, const float* hc1b,
                        const float* hc2w, const float* hc2g, const float* hc2b,
                        const float* ec2w, const float* ec2g, const float* ec2b, const float* at2,
                        const float* pdist, float* sEC1, float* aEC1, float* sHC1, float* aHC1,
                        float* sHC2, float* aHC2, float* sEC2, float* aEC2, float* hlout) {
    conv_stats_kernel<<<1458, 256, 0, stream>>>(ec1w, e0, y1buf, sEC1, B_, 128, 64, P_ * P_);
    bn_finalize_kernel<<<1, 64, 0, stream>>>(sEC1, ec1g, ec1b, aEC1, 64, invS);
    attn1_kernel<<<B_ * P_, 128, 0, stream>>>(y1buf, aEC1, at1, e1p);
    conv_stats_kernel<<<729, 256, 0, stream>>>(hc1w, e1p, tbuf, sHC1, B_, 64, 32, P_ * P_);
    bn_finalize_kernel<<<1, 32, 0, stream>>>(sHC1, hc1g, hc1b, aHC1, 32, invS);
    bn_lrelu_apply_kernel<<<729, 256, 0, stream>>>(tbuf, aHC1, 32, P_ * P_, B_ * 32 * P_ * P_);
    conv_stats_kernel<<<114, 256, 0, stream>>>(hc2w, tbuf, hlout, sHC2, B_, 32, 5, P_ * P_);
    bn_finalize_kernel<<<1, 5, 0, stream>>>(sHC2, hc2g, hc2b, aHC2, 5, invS);
    bn_lrelu_apply_kernel<<<114, 256, 0, stream>>>(hlout, aHC2, 5, P_ * P_, B_ * 5 * P_ * P_);
    conv_stats_kernel<<<1458, 256, 0, stream>>>(ec2w, e1p, y2buf, sEC2, B_, 64, 64, P_ * P_);
    bn_finalize_kernel<<<1, 64, 0, stream>>>(sEC2, ec2g, ec2b, aEC2, 64, invS);
    attn2_kernel<<<B_ * P_, 128, 0, stream>>>(y2buf, aEC2, at2, pdist, gout);
  };
  run_branch(ec1_w, ec1_g, ec1_b, at1_w, hc1_w, hc1_g, hc1_b, hc2_w, hc2_g, hc2_b,
             ec2_w, ec2_g, ec2_b, at2_w, pd,
             ws + OFF_STATS_EC1A, ws + OFF_AB_EC1A, ws + OFF_STATS_HC1A, ws + OFF_AB_HC1A,
             ws + OFF_STATS_HC2A, ws + OFF_AB_HC2A, ws + OFF_STATS_EC2A, ws + OFF_AB_EC2A,
             out_hl1);
  run_branch(ec1sp_w, ec1sp_g, ec1sp_b, at1sp_w, hc1sp_w, hc1sp_g, hc1sp_b, hc2sp_w, hc2sp_g,
             hc2sp_b, ec2sp_w, ec2sp_g, ec2sp_b, at2sp_w, pdsp,
             ws + OFF_STATS_EC1B, ws + OFF_AB_EC1B, ws + OFF_STATS_HC1B, ws + OFF_AB_HC1B,
             ws + OFF_STATS_HC2B, ws + OFF_AB_HC2B, ws + OFF_STATS_EC2B, ws + OFF_AB_EC2B,
             out_hl1sp);

  // 6. scatter parts back to points
  part2point_kernel<<<8192, 256, 0, stream>>>(gout, p2v, out0);
}